// MultiBoxLoss_5042291606172
// MI455X (gfx1250) — compile-verified
//
#include <hip/hip_runtime.h>
#include <hip/hip_bf16.h>
#include <math.h>

#define BB 64
#define PP 8732
#define CC 81
#define OO 8
#define BP (BB * PP)

typedef __attribute__((ext_vector_type(2))) float f32x2;
typedef __attribute__((ext_vector_type(8))) float f32x8;

#if defined(__has_builtin)
#if __has_builtin(__builtin_amdgcn_global_load_async_to_lds_b32) && \
    __has_builtin(__builtin_amdgcn_s_wait_asynccnt)
#define USE_ASYNC_LDS 1
#endif
#endif

typedef __attribute__((address_space(1))) int* gptr_t;   // global AS pointer
typedef __attribute__((address_space(3))) int* lptr_t;   // LDS AS pointer

__device__ __forceinline__ float smooth_l1(float x) {
    float ax = fabsf(x);
    return ax < 1.0f ? 0.5f * x * x : ax - 0.5f;
}

// ------------------------------------------------------------------
// Kernel 1: per-batch prior<->truth matching, conf_t, loc loss, pos count
// ------------------------------------------------------------------
__global__ __launch_bounds__(256) void match_kernel(
    const float* __restrict__ loc_data,   // [B,P,4]
    const float* __restrict__ targets,    // [B,O,5]
    const float* __restrict__ priors,     // [P,4] cx,cy,w,h
    int*   __restrict__ conf_t,           // [B,P]
    float* __restrict__ loss_l_b,         // [B]
    int*   __restrict__ pos_b)            // [B]
{
    __shared__ float         sbto[PP];        // best_truth_overlap
    __shared__ unsigned char sbti[PP];        // best_truth_idx
    __shared__ float sT[OO * 4];              // truth corners
    __shared__ float sL[OO];                  // labels
    __shared__ float sOv[OO][256];            // per-thread per-truth best overlap
    __shared__ int   sIx[OO][256];            // per-thread per-truth best prior idx
    __shared__ int   sBpi[OO];                // best_prior_idx
    __shared__ float sRf[256];
    __shared__ int   sRi[256];

    const int b   = blockIdx.x;
    const int tid = threadIdx.x;

    if (tid < OO * 5) {
        int o = tid / 5, c = tid - o * 5;
        float v = targets[(b * OO + o) * 5 + c];
        if (c < 4) sT[o * 4 + c] = v; else sL[o] = v;
    }
    __syncthreads();

    float bo[OO];
    int   bi[OO];
#pragma unroll
    for (int t = 0; t < OO; ++t) { bo[t] = -1.0f; bi[t] = 0x7fffffff; }

    for (int p = tid; p < PP; p += 256) {
        float pcx = priors[p * 4 + 0], pcy = priors[p * 4 + 1];
        float pw  = priors[p * 4 + 2], ph  = priors[p * 4 + 3];
        float bx1 = pcx - 0.5f * pw, by1 = pcy - 0.5f * ph;
        float bx2 = pcx + 0.5f * pw, by2 = pcy + 0.5f * ph;
        float areaB = pw * ph;
        float bestO = -1.0f; int bestT = 0;
#pragma unroll
        for (int t = 0; t < OO; ++t) {
            float ax1 = sT[t * 4 + 0], ay1 = sT[t * 4 + 1];
            float ax2 = sT[t * 4 + 2], ay2 = sT[t * 4 + 3];
            float ix = fminf(ax2, bx2) - fmaxf(ax1, bx1);
            float iy = fminf(ay2, by2) - fmaxf(ay1, by1);
            ix = fmaxf(ix, 0.0f); iy = fmaxf(iy, 0.0f);
            float inter = ix * iy;
            float areaA = (ax2 - ax1) * (ay2 - ay1);
            float iou = inter / (areaA + areaB - inter);
            if (iou > bestO) { bestO = iou; bestT = t; }   // first-occurrence argmax (axis=0)
            if (iou > bo[t]) { bo[t] = iou; bi[t] = p; }   // ascending p => first occurrence kept
        }
        sbto[p] = bestO;
        sbti[p] = (unsigned char)bestT;
    }
#pragma unroll
    for (int t = 0; t < OO; ++t) { sOv[t][tid] = bo[t]; sIx[t][tid] = bi[t]; }
    __syncthreads();

    if (tid < OO) {
        float best = -1.0f; int bidx = 0x7fffffff;
        for (int j = 0; j < 256; ++j) {
            float v = sOv[tid][j]; int ix = sIx[tid][j];
            if (v > best || (v == best && ix < bidx)) { best = v; bidx = ix; }
        }
        sBpi[tid] = bidx;
    }
    __syncthreads();

    if (tid == 0) {   // sequential t ascending => last write wins (matches scatter order)
        for (int t = 0; t < OO; ++t) {
            int pi = sBpi[t];
            sbto[pi] = 2.0f;
            sbti[pi] = (unsigned char)t;
        }
    }
    __syncthreads();

    float ll = 0.0f; int np = 0;
    for (int p = tid; p < PP; p += 256) {
        float ov = sbto[p];
        int   t  = sbti[p];
        int conf = (ov < 0.5f) ? 0 : ((int)sL[t] + 1);
        conf_t[b * PP + p] = conf;
        if (conf > 0) {
            ++np;
            float m0 = sT[t * 4 + 0], m1 = sT[t * 4 + 1];
            float m2 = sT[t * 4 + 2], m3 = sT[t * 4 + 3];
            float pcx = priors[p * 4 + 0], pcy = priors[p * 4 + 1];
            float pw  = priors[p * 4 + 2], ph  = priors[p * 4 + 3];
            float g0 = (0.5f * (m0 + m2) - pcx) / (0.1f * pw);
            float g1 = (0.5f * (m1 + m3) - pcy) / (0.1f * ph);
            float g2 = logf((m2 - m0) / pw) * 5.0f;   // /0.2
            float g3 = logf((m3 - m1) / ph) * 5.0f;
            const float* ld = loc_data + (size_t)(b * PP + p) * 4;
            ll += smooth_l1(ld[0] - g0) + smooth_l1(ld[1] - g1)
                + smooth_l1(ld[2] - g2) + smooth_l1(ld[3] - g3);
        }
    }
    sRf[tid] = ll; sRi[tid] = np;
    __syncthreads();
    for (int s = 128; s > 0; s >>= 1) {
        if (tid < s) { sRf[tid] += sRf[tid + s]; sRi[tid] += sRi[tid + s]; }
        __syncthreads();
    }
    if (tid == 0) { loss_l_b[b] = sRf[0]; pos_b[b] = sRi[0]; }
}

// ------------------------------------------------------------------
// Kernel 2: ce[row] = log(sum_k exp(conf[row,k])) - conf[row, conf_t[row]]
// One wave per 16 rows; staging via GLOBAL_LOAD_ASYNC_TO_LDS (ASYNCcnt),
// Sigma-exp via V_WMMA_F32_16X16X4_F32 x ones.
// ------------------------------------------------------------------
__global__ __launch_bounds__(256) void ce_kernel(
    const float* __restrict__ conf_data,   // [B*P, C]
    const int*   __restrict__ conf_t,      // [B*P]
    float*       __restrict__ ce)          // [B*P]
{
    __shared__ float tile[8][16 * 84];     // 84-float row stride: 8B-aligned rows,
                                           // dword stride 84 == 20 mod 64 -> conflict-free
    const int tid  = threadIdx.x;
    const int wave = tid >> 5;
    const int lane = tid & 31;
    const int rowBase = (blockIdx.x * 8 + wave) * 16;

    // Stage 16 rows x 81 classes = 1296 contiguous floats, fully coalesced.
    const float* src = conf_data + (size_t)rowBase * CC;
#ifdef USE_ASYNC_LDS
    // Direct global->LDS copy, no VGPR round-trip; tracked on ASYNCcnt.
    for (int i = lane; i < 16 * CC; i += 32) {
        int r = i / CC;
        int c = i - r * CC;
        __builtin_amdgcn_global_load_async_to_lds_b32(
            (gptr_t)(src + i),
            (lptr_t)&tile[wave][r * 84 + c],
            0, 0);
    }
    // Each wave consumes only its own tile -> wave-local wait, no block barrier.
    __builtin_amdgcn_s_wait_asynccnt(0);
#else
    for (int i = lane; i < 16 * CC; i += 32) {
        int r = i / CC;
        int c = i - r * CC;
        tile[wave][r * 84 + c] = src[i];
    }
    __syncthreads();
#endif

    // A-fragment layout for 16x4 f32 (ISA 7.12.2):
    // lanes 0-15 : row M = lane,   VGPR0=K0, VGPR1=K1
    // lanes 16-31: row M = lane-16,VGPR0=K2, VGPR1=K3
    const int rowL = lane & 15;
    const int hi   = lane >> 4;
    const float* myrow = &tile[wave][rowL * 84];

    f32x8 acc = {};
    f32x2 ones; ones.x = 1.0f; ones.y = 1.0f;   // B = 4x16 all-ones
    for (int c = 0; c < 21; ++c) {              // 21 chunks of K=4 cover 81 (pad 0)
        int k = c * 4 + hi * 2;
        f32x2 a;
        a.x = (k     < CC) ? __expf(myrow[k])     : 0.0f;
        a.y = (k + 1 < CC) ? __expf(myrow[k + 1]) : 0.0f;
        // D[m,n] = sum_k A[m,k]*1 + C[m,n] -> every column holds row sums
        acc = __builtin_amdgcn_wmma_f32_16x16x4_f32(
            false, a, false, ones, (short)0, acc, false, false);
    }

    // D layout: VGPR r, lane 0 -> (M=r, N=0); lane 16 -> (M=8+r, N=0).
    if (lane == 0 || lane == 16) {
#pragma unroll
        for (int r = 0; r < 8; ++r) {
            int rloc = hi * 8 + r;
            int gRow = rowBase + rloc;
            float s   = acc[r];
            float lse = logf(s);
            int   ct  = conf_t[gRow];
            float tl  = tile[wave][rloc * 84 + ct];
            ce[gRow]  = lse - tl;
        }
    }
}

// ------------------------------------------------------------------
// Kernel 3: per-batch hard-negative mining via exact radix k-th select
// ------------------------------------------------------------------
__global__ __launch_bounds__(256) void mine_kernel(
    const float* __restrict__ ce,      // [B,P]
    const int*   __restrict__ conf_t,  // [B,P]
    const int*   __restrict__ pos_b,   // [B]
    float*       __restrict__ loss_c_b)// [B]
{
    __shared__ unsigned int sm[PP];    // mined as IEEE bits (all >= 0)
    __shared__ float sRf[256];
    __shared__ int   sRi[256];

    const int b = blockIdx.x, tid = threadIdx.x;

    float posSum = 0.0f;
    for (int p = tid; p < PP; p += 256) {
        float v  = ce[b * PP + p];
        int   ct = conf_t[b * PP + p];
        bool pos = ct > 0;
        if (pos) posSum += v;
        float mined = pos ? 0.0f : fmaxf(v, 0.0f);
        sm[p] = __float_as_uint(mined);
    }
    sRf[tid] = posSum;
    __syncthreads();
    for (int s = 128; s > 0; s >>= 1) {
        if (tid < s) sRf[tid] += sRf[tid + s];
        __syncthreads();
    }
    float posTot = sRf[0];
    __syncthreads();

    int np = pos_b[b];
    int k  = 3 * np; if (k > PP - 1) k = PP - 1;   // num_neg (integer-valued)

    float negTerm = 0.0f;
    if (k > 0) {
        // v = k-th largest mined value: max x with countGE(x) >= k (monotone bits)
        unsigned int prefix = 0u;
        for (int bit = 31; bit >= 0; --bit) {
            unsigned int cand = prefix | (1u << bit);
            int cnt = 0;
            for (int p = tid; p < PP; p += 256) cnt += (sm[p] >= cand) ? 1 : 0;
            sRi[tid] = cnt;
            __syncthreads();
            for (int s = 128; s > 0; s >>= 1) {
                if (tid < s) sRi[tid] += sRi[tid + s];
                __syncthreads();
            }
            int total = sRi[0];
            __syncthreads();
            if (total >= k) prefix = cand;
        }
        // sum of top-k = sum(mined > v) + (k - count(mined > v)) * v
        int cg = 0; float sg = 0.0f;
        for (int p = tid; p < PP; p += 256) {
            unsigned int u = sm[p];
            if (u > prefix) { ++cg; sg += __uint_as_float(u); }
        }
        sRi[tid] = cg; sRf[tid] = sg;
        __syncthreads();
        for (int s = 128; s > 0; s >>= 1) {
            if (tid < s) { sRi[tid] += sRi[tid + s]; sRf[tid] += sRf[tid + s]; }
            __syncthreads();
        }
        negTerm = sRf[0] + ((float)k - (float)sRi[0]) * __uint_as_float(prefix);
    }
    if (tid == 0) loss_c_b[b] = posTot + negTerm;
}

// ------------------------------------------------------------------
// Kernel 4: fixed-order scalar combine
// ------------------------------------------------------------------
__global__ void final_kernel(const float* __restrict__ loss_l_b,
                             const float* __restrict__ loss_c_b,
                             const int*   __restrict__ pos_b,
                             float*       __restrict__ out)
{
    if (threadIdx.x == 0 && blockIdx.x == 0) {
        float ll = 0.0f, lc = 0.0f; int n = 0;
        for (int b = 0; b < BB; ++b) { ll += loss_l_b[b]; lc += loss_c_b[b]; n += pos_b[b]; }
        float N = (float)n;
        out[0] = ll / N + lc / N;
    }
}

extern "C" void kernel_launch(void* const* d_in, const int* in_sizes, int n_in,
                              void* d_out, int out_size, void* d_ws, size_t ws_size,
                              hipStream_t stream) {
    const float* loc    = (const float*)d_in[0];
    const float* conf   = (const float*)d_in[1];
    const float* targ   = (const float*)d_in[2];
    const float* priors = (const float*)d_in[3];

    // workspace layout: conf_t [BP] int | ce [BP] float | ll_b [B] | lc_b [B] | pos_b [B]
    int*   conf_t = (int*)d_ws;
    float* ce     = (float*)((char*)d_ws + (size_t)BP * 4);
    float* ll_b   = (float*)((char*)d_ws + (size_t)BP * 8);
    float* lc_b   = ll_b + BB;
    int*   pos_b  = (int*)(lc_b + BB);

    match_kernel<<<BB, 256, 0, stream>>>(loc, targ, priors, conf_t, ll_b, pos_b);
    ce_kernel<<<BP / 128, 256, 0, stream>>>(conf, conf_t, ce);   // BP = 558848 = 4366*128 exactly
    mine_kernel<<<BB, 256, 0, stream>>>(ce, conf_t, pos_b, lc_b);
    final_kernel<<<1, 64, 0, stream>>>(ll_b, lc_b, pos_b, (float*)d_out);
}